// SyncGraph_88038239633884
// MI455X (gfx1250) — compile-verified
//
#include <hip/hip_runtime.h>

typedef float v2f __attribute__((ext_vector_type(2)));
typedef float v8f __attribute__((ext_vector_type(8)));

#define NB 128
#define NP 32
#define ND 256
#define DIST_THD 1.5f

// ---------------------------------------------------------------------------
// WMMA helpers: fp32 16x16x4, D = A(16xK) * B(Kx16) + C
// A fragment (ISA 7.12.2, 32-bit A 16x4): lane<16 -> M=lane, k+{0,1};
//                                         lane>=16 -> M=lane-16, k+{2,3}
// B fragment mirrored: lane<16 -> N=lane, rows k+{0,1}; lane>=16 rows k+{2,3}
// D: VGPR r -> M = m0 + r + (lane>=16 ? 8 : 0), N = n0 + (lane&15)
// ---------------------------------------------------------------------------
__device__ __forceinline__ v8f mm_AB(const float* __restrict__ A, int lda, int m0,
                                     const float* __restrict__ Bm, int ldb, int n0,
                                     int K, int lane, v8f acc)
{
    const int lh   = lane & 15;
    const int koff = (lane >> 4) << 1;           // 0 or 2
    const float* arow = A + (m0 + lh) * lda;
    const float* bcol = Bm + n0 + lh;
    for (int k = 0; k < K; k += 4) {
        v2f a, b;
        a.x = arow[k + koff];
        a.y = arow[k + koff + 1];
        b.x = bcol[(k + koff) * ldb];
        b.y = bcol[(k + koff + 1) * ldb];
        acc = __builtin_amdgcn_wmma_f32_16x16x4_f32(false, a, false, b,
                                                    (short)0, acc, false, false);
    }
    return acc;
}

// B given transposed (row-major [N,K]) e.g. diff_wts
__device__ __forceinline__ v8f mm_ABt(const float* __restrict__ A, int lda, int m0,
                                      const float* __restrict__ Bt, int ldbt, int n0,
                                      int K, int lane, v8f acc)
{
    const int lh   = lane & 15;
    const int koff = (lane >> 4) << 1;
    const float* arow = A + (m0 + lh) * lda;
    const float* brow = Bt + (n0 + lh) * ldbt;
    for (int k = 0; k < K; k += 4) {
        v2f a, b;
        a.x = arow[k + koff];
        a.y = arow[k + koff + 1];
        b.x = brow[k + koff];
        b.y = brow[k + koff + 1];
        acc = __builtin_amdgcn_wmma_f32_16x16x4_f32(false, a, false, b,
                                                    (short)0, acc, false, false);
    }
    return acc;
}

__device__ __forceinline__ void store_tile(float* __restrict__ dst, int ldd,
                                           int m0, int n0, int lane, v8f acc,
                                           const float* __restrict__ bias,
                                           float biasScale, int doRelu)
{
    const int lh    = lane & 15;
    const int rbase = m0 + ((lane >> 4) << 3);
    const int col   = n0 + lh;
    float bv = bias ? bias[col] * biasScale : 0.f;
#pragma unroll
    for (int r = 0; r < 8; ++r) {
        float v = acc[r] + bv;
        if (doRelu) v = fmaxf(v, 0.f);
        dst[(rbase + r) * ldd + col] = v;
    }
}

__device__ __forceinline__ void ldg_lds(float* __restrict__ dst,
                                        const float* __restrict__ src,
                                        int nFloats, int tid, int nThr)
{
    const float4* s4 = (const float4*)src;
    float4*       d4 = (float4*)dst;
    for (int i = tid; i < (nFloats >> 2); i += nThr) d4[i] = s4[i];
}

// ---------------------------------------------------------------------------
// K1: a = fea@W1[0:256,:], c = fea@W1[256:512,:]   (per batch)
// ---------------------------------------------------------------------------
__global__ void k_ac(const float* __restrict__ fea, const float* __restrict__ w1,
                     float* __restrict__ bufA, float* __restrict__ bufC)
{
    __shared__ float F[NP * ND];
    const int b = blockIdx.x, tid = threadIdx.x;
    const int lane = tid & 31, wave = tid >> 5;
    ldg_lds(F, fea + (size_t)b * NP * ND, NP * ND, tid, blockDim.x);
    __syncthreads();
    const float* W1a = w1;
    const float* W1c = w1 + ND * ND;
    float* oa = bufA + (size_t)b * NP * ND;
    float* oc = bufC + (size_t)b * NP * ND;
    for (int t = wave; t < 2 * (ND / 16); t += 8) {
        const int m0 = (t & 1) * 16, n0 = (t >> 1) * 16;
        v8f acc = {0.f,0.f,0.f,0.f,0.f,0.f,0.f,0.f};
        acc = mm_AB(F, ND, m0, W1a, ND, n0, ND, lane, acc);
        store_tile(oa, ND, m0, n0, lane, acc, nullptr, 0.f, 0);
        v8f acc2 = {0.f,0.f,0.f,0.f,0.f,0.f,0.f,0.f};
        acc2 = mm_AB(F, ND, m0, W1c, ND, n0, ND, lane, acc2);
        store_tile(oc, ND, m0, n0, lane, acc2, nullptr, 0.f, 0);
    }
}

// ---------------------------------------------------------------------------
// K2: presum[p,d] = sum_q relu(a[p,d] + c[q,d] + b1[d])   (in place over a)
// ---------------------------------------------------------------------------
__global__ void k_presum(float* __restrict__ bufA, const float* __restrict__ bufC,
                         const float* __restrict__ b1)
{
    __shared__ float C[NP * ND];
    const int b = blockIdx.x, tid = threadIdx.x;
    ldg_lds(C, bufC + (size_t)b * NP * ND, NP * ND, tid, blockDim.x);
    __syncthreads();
    const int d = tid;                       // 256 threads == ND
    const float b1v = b1[d];
    float* arow = bufA + (size_t)b * NP * ND;
    for (int p = 0; p < NP; ++p) {
        const float av = arow[p * ND + d] + b1v;
        float s = 0.f;
#pragma unroll
        for (int q = 0; q < NP; ++q) s += fmaxf(av + C[q * ND + d], 0.f);
        arow[p * ND + d] = s;
    }
}

// ---------------------------------------------------------------------------
// K3/K5: out = in @ W [256x256] + biasScale*bias (opt relu)
// ---------------------------------------------------------------------------
__global__ void k_gemm256(const float* __restrict__ in, const float* __restrict__ W,
                          const float* __restrict__ bias, float biasScale, int doRelu,
                          float* __restrict__ out)
{
    __shared__ float A[NP * ND];
    const int b = blockIdx.x, tid = threadIdx.x;
    const int lane = tid & 31, wave = tid >> 5;
    ldg_lds(A, in + (size_t)b * NP * ND, NP * ND, tid, blockDim.x);
    __syncthreads();
    float* o = out + (size_t)b * NP * ND;
    for (int t = wave; t < 2 * (ND / 16); t += 8) {
        const int m0 = (t & 1) * 16, n0 = (t >> 1) * 16;
        v8f acc = {0.f,0.f,0.f,0.f,0.f,0.f,0.f,0.f};
        acc = mm_AB(A, ND, m0, W, ND, n0, ND, lane, acc);
        store_tile(o, ND, m0, n0, lane, acc, bias, biasScale, doRelu);
    }
}

// ---------------------------------------------------------------------------
// K4/K7: out = [in1,in2] @ W [512x256] (+bias) (opt relu)   K split 256+256
// ---------------------------------------------------------------------------
__global__ void k_gemm2(const float* __restrict__ in1, const float* __restrict__ in2,
                        const float* __restrict__ W, const float* __restrict__ bias,
                        int doRelu, float* __restrict__ out)
{
    __shared__ float A1[NP * ND];
    __shared__ float A2[NP * ND];
    const int b = blockIdx.x, tid = threadIdx.x;
    const int lane = tid & 31, wave = tid >> 5;
    ldg_lds(A1, in1 + (size_t)b * NP * ND, NP * ND, tid, blockDim.x);
    ldg_lds(A2, in2 + (size_t)b * NP * ND, NP * ND, tid, blockDim.x);
    __syncthreads();
    float* o = out + (size_t)b * NP * ND;
    for (int t = wave; t < 2 * (ND / 16); t += 8) {
        const int m0 = (t & 1) * 16, n0 = (t >> 1) * 16;
        v8f acc = {0.f,0.f,0.f,0.f,0.f,0.f,0.f,0.f};
        acc = mm_AB(A1, ND, m0, W,            ND, n0, ND, lane, acc);
        acc = mm_AB(A2, ND, m0, W + ND * ND,  ND, n0, ND, lane, acc);
        store_tile(o, ND, m0, n0, lane, acc, bias, 1.f, doRelu);
    }
}

// ---------------------------------------------------------------------------
// K6: softmax-weighted residual diff + normalized adjacency
// ---------------------------------------------------------------------------
__global__ void k_diff(const float* __restrict__ fea, const float* __restrict__ pos,
                       const float* __restrict__ dw, const float* __restrict__ dbias,
                       float* __restrict__ outDiff, float* __restrict__ adjW)
{
    __shared__ float F[NP * ND];
    __shared__ float L[NP * NP];
    __shared__ float A4[NP * NP];
    __shared__ float dinv[NP];
    __shared__ float p2[NP * 2];
    const int b = blockIdx.x, tid = threadIdx.x;
    const int lane = tid & 31, wave = tid >> 5;
    ldg_lds(F, fea + (size_t)b * NP * ND, NP * ND, tid, blockDim.x);
    if (tid < NP * 2) p2[tid] = pos[(size_t)b * NP * 2 + tid];
    __syncthreads();

    if (wave < 4) {
        // logits[p,q] = fea[p,:] . dw[q,:] + dbias[q]   (4 tiles, one per wave)
        const int m0 = (wave & 1) * 16, n0 = (wave >> 1) * 16;
        v8f acc = {0.f,0.f,0.f,0.f,0.f,0.f,0.f,0.f};
        acc = mm_ABt(F, ND, m0, dw, ND, n0, ND, lane, acc);
        store_tile(L, NP, m0, n0, lane, acc, dbias, 1.f, 0);
    } else {
        // adjacency (with self-loop) from positions
        for (int idx = tid - 128; idx < NP * NP; idx += 128) {
            const int p = idx >> 5, q = idx & 31;
            const float dx = p2[p * 2] - p2[q * 2];
            const float dy = p2[p * 2 + 1] - p2[q * 2 + 1];
            const float dist = sqrtf(dx * dx + dy * dy);
            A4[idx] = (p == q) ? 1.f : ((dist < DIST_THD) ? 1.f : 0.f);
        }
    }
    __syncthreads();

    if (tid < NP) {                 // row softmax
        float m = -3.4e38f;
        for (int q = 0; q < NP; ++q) m = fmaxf(m, L[tid * NP + q]);
        float s = 0.f;
        for (int q = 0; q < NP; ++q) { float e = expf(L[tid * NP + q] - m); L[tid * NP + q] = e; s += e; }
        const float inv = 1.f / s;
        for (int q = 0; q < NP; ++q) L[tid * NP + q] *= inv;
    } else if (tid < 2 * NP) {      // degree^-1/2
        const int p = tid - NP;
        float s = 1e-6f;
        for (int q = 0; q < NP; ++q) s += A4[p * NP + q];
        dinv[p] = 1.f / sqrtf(s);
    }
    __syncthreads();

    // diff = fea - w @ fea   (K = 32)
    float* od = outDiff + (size_t)b * NP * ND;
    for (int t = wave; t < 2 * (ND / 16); t += 8) {
        const int m0 = (t & 1) * 16, n0 = (t >> 1) * 16;
        v8f acc = {0.f,0.f,0.f,0.f,0.f,0.f,0.f,0.f};
        acc = mm_AB(L, NP, m0, F, ND, n0, NP, lane, acc);
        const int lh = lane & 15, rbase = m0 + ((lane >> 4) << 3), col = n0 + lh;
#pragma unroll
        for (int r = 0; r < 8; ++r)
            od[(rbase + r) * ND + col] = F[(rbase + r) * ND + col] - acc[r];
    }
    // norm_adj = dinv * A * dinv^T
    for (int idx = tid; idx < NP * NP; idx += blockDim.x)
        adjW[(size_t)b * NP * NP + idx] = dinv[idx >> 5] * A4[idx] * dinv[idx & 31];
}

// ---------------------------------------------------------------------------
// K8: out = relu(norm_adj @ t)   (t staged from d_out into LDS, in-place)
// ---------------------------------------------------------------------------
__global__ void k_out(const float* __restrict__ tbuf, const float* __restrict__ adjW,
                      float* __restrict__ out)
{
    __shared__ float T[NP * ND];
    __shared__ float A4[NP * NP];
    const int b = blockIdx.x, tid = threadIdx.x;
    const int lane = tid & 31, wave = tid >> 5;
    ldg_lds(T, tbuf + (size_t)b * NP * ND, NP * ND, tid, blockDim.x);
    ldg_lds(A4, adjW + (size_t)b * NP * NP, NP * NP, tid, blockDim.x);
    __syncthreads();
    float* o = out + (size_t)b * NP * ND;
    for (int t = wave; t < 2 * (ND / 16); t += 8) {
        const int m0 = (t & 1) * 16, n0 = (t >> 1) * 16;
        v8f acc = {0.f,0.f,0.f,0.f,0.f,0.f,0.f,0.f};
        acc = mm_AB(A4, NP, m0, T, ND, n0, NP, lane, acc);
        store_tile(o, ND, m0, n0, lane, acc, nullptr, 0.f, 1);
    }
}

// ---------------------------------------------------------------------------
extern "C" void kernel_launch(void* const* d_in, const int* in_sizes, int n_in,
                              void* d_out, int out_size, void* d_ws, size_t ws_size,
                              hipStream_t stream)
{
    const float* fea   = (const float*)d_in[0];
    const float* pos   = (const float*)d_in[1];
    const float* pe_w1 = (const float*)d_in[2];
    const float* pe_b1 = (const float*)d_in[3];
    const float* pe_w2 = (const float*)d_in[4];
    const float* pe_b2 = (const float*)d_in[5];
    const float* sm_w1 = (const float*)d_in[6];
    const float* sm_b1 = (const float*)d_in[7];
    const float* sm_w2 = (const float*)d_in[8];
    const float* sm_b2 = (const float*)d_in[9];
    const float* dwts  = (const float*)d_in[10];
    const float* dbias = (const float*)d_in[11];
    const float* gconv = (const float*)d_in[12];
    float* out = (float*)d_out;

    const size_t NBPD = (size_t)NB * NP * ND;       // 1,048,576 floats
    float* bufA = (float*)d_ws;                     // [B,P,D]
    float* bufC = bufA + NBPD;                      // [B,P,D]
    float* adjW = bufC + NBPD;                      // [B,P,P]

    dim3 grid(NB), blk(256);
    // 1) a,c
    k_ac<<<grid, blk, 0, stream>>>(fea, pe_w1, bufA, bufC);
    // 2) presum (in place in bufA)
    k_presum<<<grid, blk, 0, stream>>>(bufA, bufC, pe_b1);
    // 3) interact = presum@pe_w2 + P*pe_b2 -> bufC
    k_gemm256<<<grid, blk, 0, stream>>>(bufA, pe_w2, pe_b2, (float)NP, 0, bufC);
    // 4) sh = relu([fea,interact]@sm_w1 + sm_b1) -> bufA
    k_gemm2<<<grid, blk, 0, stream>>>(fea, bufC, sm_w1, sm_b1, 1, bufA);
    // 5) spatial = sh@sm_w2 + sm_b2 -> bufC
    k_gemm256<<<grid, blk, 0, stream>>>(bufA, sm_w2, sm_b2, 1.f, 0, bufC);
    // 6) diff -> bufA ; norm_adj -> adjW
    k_diff<<<grid, blk, 0, stream>>>(fea, pos, dwts, dbias, bufA, adjW);
    // 7) t = [spatial,diff]@gconv_wt -> d_out   (reassociated graph conv)
    k_gemm2<<<grid, blk, 0, stream>>>(bufC, bufA, gconv, nullptr, 0, out);
    // 8) out = relu(norm_adj @ t) in place (LDS-staged)
    k_out<<<grid, blk, 0, stream>>>(out, adjW, out);
}